// NeuralLogicProgramming_81587198755486
// MI455X (gfx1250) — compile-verified
//
#include <hip/hip_runtime.h>

#define N_ENT  15000
#define N_RELS 237
#define R2     474
#define HID    128
#define NSTEP  3
#define EBASE  200000
#define E2     400000
#define BN     128      // batch columns after unique()
#define GATES  512      // 4*HID

typedef float v2f __attribute__((ext_vector_type(2)));
typedef float v8f __attribute__((ext_vector_type(8)));

__device__ __forceinline__ float sigmoidf_(float x) { return 1.0f / (1.0f + expf(-x)); }

// ---------------------------------------------------------------------------
// Preprocess: negative_sample_to_tail + unique(sorted, size=128, fill=0)
// ---------------------------------------------------------------------------
__global__ void prep_kernel(const int* __restrict__ h_index,
                            const int* __restrict__ t_index,
                            const int* __restrict__ r_index,
                            int* __restrict__ h_set, int* __restrict__ r_set,
                            int* __restrict__ inv_out, int* __restrict__ t_sel) {
    __shared__ int hsh[BN], tsh[BN], hr[BN], firsto[BN], invsh[BN];
    int j = threadIdx.x;
    if (j < BN) { hsh[j] = h_index[j]; tsh[j] = t_index[j]; }
    __syncthreads();
    if (j < BN) {
        int row = j >> 4;
        int first = hsh[row * 16];
        bool all_eq = true;
        for (int c = 0; c < 16; ++c) all_eq = all_eq && (hsh[row * 16 + c] == first);
        int hi = all_eq ? hsh[j] : tsh[j];
        int ti = all_eq ? tsh[j] : hsh[j];
        int ri = all_eq ? r_index[j] : (r_index[j] + N_RELS);
        hr[j] = hi * R2 + ri;
        t_sel[j] = ti;
    }
    __syncthreads();
    if (j < BN) {
        bool fo = true;
        for (int m = 0; m < j; ++m) if (hr[m] == hr[j]) { fo = false; break; }
        firsto[j] = fo ? 1 : 0;
    }
    __syncthreads();
    if (j < BN) {
        int pos = 0, x = hr[j];
        for (int k = 0; k < BN; ++k) pos += (firsto[k] && (hr[k] < x)) ? 1 : 0;
        invsh[j] = pos;
        inv_out[j] = pos;
    }
    __syncthreads();
    if (j < BN) {
        int u = 0;  // padding slots keep fill_value 0
        for (int k = 0; k < BN; ++k)
            if (firsto[k] && invsh[k] == j) u = hr[k];
        h_set[j] = u / R2;
        r_set[j] = u % R2;
    }
}

// ---------------------------------------------------------------------------
// LSTM: single block, 8 waves; WMMA f32 16x16x4 GEMMs, serial over T
// ---------------------------------------------------------------------------
__global__ void lstm_kernel(const float* __restrict__ qemb,
                            const float* __restrict__ wih,
                            const float* __restrict__ whh,
                            const float* __restrict__ bih,
                            const float* __restrict__ bhh,
                            const int* __restrict__ r_set,
                            float* __restrict__ hstate, float* __restrict__ cstate,
                            float* __restrict__ hidden, float* __restrict__ gates) {
    const int tid  = threadIdx.x;
    const int wave = tid >> 5;
    const int lane = tid & 31;
    const int lrow = lane & 15;
    const int khi  = (lane >> 4) * 2;  // A/B K sub-offset: 0 or 2
    const int chi  = (lane >> 4) * 8;  // C/D row sub-offset: 0 or 8

    for (int idx = tid; idx < BN * HID; idx += blockDim.x) {
        hstate[idx] = 0.0f; cstate[idx] = 0.0f;
    }
    __threadfence();
    __syncthreads();

    for (int t = 0; t < NSTEP; ++t) {
        // gates[b][g] = X[b]·Wih[g] + H[b]·Whh[g] + bias[g]
        for (int tile = wave; tile < 256; tile += 8) {
            int mt = tile & 7, nt = tile >> 3;
            int m0 = mt * 16, n0 = nt * 16;
            int arow = m0 + lrow;
            int q = (t < NSTEP - 1) ? r_set[arow] : R2;        // end token
            const float* xrow = qemb + (long)q * HID;
            const float* hrow = hstate + (long)arow * HID;
            int bcol = n0 + lrow;
            v8f acc = {0.f, 0.f, 0.f, 0.f, 0.f, 0.f, 0.f, 0.f};
            for (int k0 = 0; k0 < HID; k0 += 4) {
                v2f a, b;
                a[0] = xrow[k0 + khi];     a[1] = xrow[k0 + khi + 1];
                b[0] = wih[bcol * HID + k0 + khi];
                b[1] = wih[bcol * HID + k0 + khi + 1];
                acc = __builtin_amdgcn_wmma_f32_16x16x4_f32(false, a, false, b,
                                                            (short)0, acc, false, false);
            }
            for (int k0 = 0; k0 < HID; k0 += 4) {
                v2f a, b;
                a[0] = hrow[k0 + khi];     a[1] = hrow[k0 + khi + 1];
                b[0] = whh[bcol * HID + k0 + khi];
                b[1] = whh[bcol * HID + k0 + khi + 1];
                acc = __builtin_amdgcn_wmma_f32_16x16x4_f32(false, a, false, b,
                                                            (short)0, acc, false, false);
            }
            float bias = bih[bcol] + bhh[bcol];
            for (int v = 0; v < 8; ++v)
                gates[(m0 + v + chi) * GATES + bcol] = acc[v] + bias;
        }
        __threadfence();
        __syncthreads();
        // elementwise gate update (i, f, g, o)
        for (int idx = tid; idx < BN * HID; idx += blockDim.x) {
            int b = idx >> 7, hc = idx & 127;
            float gi = gates[b * GATES + hc];
            float gf = gates[b * GATES + HID + hc];
            float gg = gates[b * GATES + 2 * HID + hc];
            float go = gates[b * GATES + 3 * HID + hc];
            float c = sigmoidf_(gf) * cstate[idx] + sigmoidf_(gi) * tanhf(gg);
            float h = sigmoidf_(go) * tanhf(c);
            cstate[idx] = c;
            hstate[idx] = h;
            hidden[t * BN * HID + idx] = h;
        }
        __threadfence();
        __syncthreads();
    }
}

// ---------------------------------------------------------------------------
// logits[b][r] = hidden[step][b]·weight_w[r] + weight_b[r], via WMMA (N pad 480)
// ---------------------------------------------------------------------------
__global__ void wlog_kernel(const float* __restrict__ hidden,
                            const float* __restrict__ weight_w,
                            const float* __restrict__ weight_b,
                            float* __restrict__ wlog, int step) {
    const float* kmat = hidden + step * BN * HID;
    int tid = threadIdx.x;
    int wave = tid >> 5, lane = tid & 31;
    int gtile = blockIdx.x * 8 + wave;          // 240 tiles: 8 M x 30 N
    if (gtile >= 240) return;
    int mt = gtile & 7, nt = gtile >> 3;
    int m0 = mt * 16, n0 = nt * 16;
    int lrow = lane & 15;
    int khi = (lane >> 4) * 2;
    int chi = (lane >> 4) * 8;
    const float* arow = kmat + (m0 + lrow) * HID;
    int bcol = n0 + lrow;
    bool bvalid = bcol < R2;
    v8f acc = {0.f, 0.f, 0.f, 0.f, 0.f, 0.f, 0.f, 0.f};
    for (int k0 = 0; k0 < HID; k0 += 4) {
        v2f a, b;
        a[0] = arow[k0 + khi]; a[1] = arow[k0 + khi + 1];
        b[0] = bvalid ? weight_w[bcol * HID + k0 + khi] : 0.0f;
        b[1] = bvalid ? weight_w[bcol * HID + k0 + khi + 1] : 0.0f;
        acc = __builtin_amdgcn_wmma_f32_16x16x4_f32(false, a, false, b,
                                                    (short)0, acc, false, false);
    }
    if (bvalid) {
        float bias = weight_b[bcol];
        for (int v = 0; v < 8; ++v)
            wlog[(m0 + v + chi) * R2 + bcol] = acc[v] + bias;
    }
}

// softmax over R2 per batch column; write transposed wsoft[r*BN + b]
__global__ void wsoftmax_kernel(const float* __restrict__ wlog,
                                float* __restrict__ wsoft) {
    int b = blockIdx.x, tid = threadIdx.x;
    __shared__ float red[256];
    const float* row = wlog + b * R2;
    float mx = -1e30f;
    for (int r = tid; r < R2; r += 256) mx = fmaxf(mx, row[r]);
    red[tid] = mx; __syncthreads();
    for (int s = 128; s > 0; s >>= 1) { if (tid < s) red[tid] = fmaxf(red[tid], red[tid + s]); __syncthreads(); }
    mx = red[0]; __syncthreads();
    float sum = 0.0f;
    for (int r = tid; r < R2; r += 256) sum += expf(row[r] - mx);
    red[tid] = sum; __syncthreads();
    for (int s = 128; s > 0; s >>= 1) { if (tid < s) red[tid] += red[tid + s]; __syncthreads(); }
    float inv = 1.0f / red[0];
    for (int r = tid; r < R2; r += 256) wsoft[r * BN + b] = expf(row[r] - mx) * inv;
}

// attn over history steps (<=3 terms), per batch column
__global__ void attn_kernel(const float* __restrict__ hidden,
                            float* __restrict__ attn, int step) {
    int b = threadIdx.x;
    if (b >= BN) return;
    const float* k = hidden + step * BN * HID + b * HID;
    float s[NSTEP];
    float mx = -1e30f;
    for (int t = 0; t <= step; ++t) {
        const float* h = hidden + t * BN * HID + b * HID;
        float d = 0.0f;
        for (int j = 0; j < HID; ++j) d += k[j] * h[j];
        s[t] = d; mx = fmaxf(mx, d);
    }
    float sum = 0.0f;
    for (int t = 0; t <= step; ++t) { s[t] = expf(s[t] - mx); sum += s[t]; }
    for (int t = 0; t <= step; ++t) attn[t * BN + b] = s[t] / sum;
}

// inp[n,b] = attn0*onehot(h_set) + attn1*out0 + attn2*out1
__global__ void inp_kernel(const float* __restrict__ attn,
                           const int* __restrict__ h_set,
                           const float* __restrict__ out0,
                           const float* __restrict__ out1,
                           float* __restrict__ inp, int step) {
    int idx = blockIdx.x * blockDim.x + threadIdx.x;
    if (idx >= N_ENT * BN) return;
    int n = idx >> 7, b = idx & 127;
    float v = (h_set[b] == n) ? attn[b] : 0.0f;
    if (step >= 1) v += attn[BN + b] * out0[idx];
    if (step >= 2) v += attn[2 * BN + b] * out1[idx];
    inp[idx] = v;
}

__global__ void zero_kernel(float* __restrict__ p, int n) {
    int i = blockIdx.x * blockDim.x + threadIdx.x;
    if (i < n) p[i] = 0.0f;
}

// one wave per (directed) edge; 128 contiguous float atomics per edge
__global__ void edge_kernel(const int* __restrict__ src, const int* __restrict__ dst,
                            const int* __restrict__ rel, const float* __restrict__ ew,
                            const float* __restrict__ inp, const float* __restrict__ wsoft,
                            float* __restrict__ out) {
    int gw = (blockIdx.x * blockDim.x + threadIdx.x) >> 5;
    int lane = threadIdx.x & 31;
    if (gw >= E2) return;
    int nin, nout, r;
    if (gw < EBASE) { nin = src[gw]; nout = dst[gw]; r = rel[gw]; }
    else { int e0 = gw - EBASE; nin = dst[e0]; nout = src[e0]; r = rel[e0] + N_RELS; }
    float w = ew[gw];
    const float* ip = inp + (long)nin * BN;
    const float* wp = wsoft + (long)r * BN;
    float* op = out + (long)nout * BN;
    for (int j = 0; j < 4; ++j) {
        int b = lane + 32 * j;
        atomicAdd(op + b, ip[b] * wp[b] * w);
    }
}

// per-column sum + scale (fused)
__global__ void norm_kernel(float* __restrict__ out) {
    int b = blockIdx.x, tid = threadIdx.x;
    __shared__ float red[256];
    float s = 0.0f;
    for (int n = tid; n < N_ENT; n += 256) s += out[n * BN + b];
    red[tid] = s; __syncthreads();
    for (int st = 128; st > 0; st >>= 1) { if (tid < st) red[tid] += red[tid + st]; __syncthreads(); }
    float inv = 1.0f / fmaxf(red[0], 1e-10f);
    for (int n = tid; n < N_ENT; n += 256) out[n * BN + b] *= inv;
}

__global__ void score_kernel(const float* __restrict__ out2,
                             const int* __restrict__ t_sel,
                             const int* __restrict__ inv_idx,
                             const float* __restrict__ lw,
                             const float* __restrict__ lb,
                             float* __restrict__ d_out) {
    int j = threadIdx.x;
    if (j < BN) d_out[j] = out2[(long)t_sel[j] * BN + inv_idx[j]] * lw[0] + lb[0];
}

// ---------------------------------------------------------------------------
extern "C" void kernel_launch(void* const* d_in, const int* in_sizes, int n_in,
                              void* d_out, int out_size, void* d_ws, size_t ws_size,
                              hipStream_t stream) {
    const float* qemb     = (const float*)d_in[0];
    const float* wih      = (const float*)d_in[1];
    const float* whh      = (const float*)d_in[2];
    const float* bih      = (const float*)d_in[3];
    const float* bhh      = (const float*)d_in[4];
    const float* weight_w = (const float*)d_in[5];
    const float* weight_b = (const float*)d_in[6];
    const float* lin_w    = (const float*)d_in[7];
    const float* lin_b    = (const float*)d_in[8];
    const float* ew       = (const float*)d_in[9];
    const int*   h_index  = (const int*)d_in[10];
    const int*   t_index  = (const int*)d_in[11];
    const int*   r_index  = (const int*)d_in[12];
    const int*   src      = (const int*)d_in[13];
    const int*   dst      = (const int*)d_in[14];
    const int*   rel      = (const int*)d_in[15];

    char* ws = (char*)d_ws;
    int*   h_set  = (int*)(ws + 0);
    int*   r_set  = (int*)(ws + 512);
    int*   inv_i  = (int*)(ws + 1024);
    int*   t_sel  = (int*)(ws + 1536);
    float* attn   = (float*)(ws + 2048);                 // 3*128
    float* hidden = (float*)(ws + 4096);                 // 3*128*128
    float* hstate = (float*)(ws + 200704);               // 128*128
    float* cstate = (float*)(ws + 266240);               // 128*128
    float* gates  = (float*)(ws + 331776);               // 128*512
    float* wlog   = (float*)(ws + 593920);               // 128*474
    float* wsoft  = (float*)(ws + 836608);               // 474*128
    float* inp    = (float*)(ws + 1079296);              // 15000*128
    float* out0   = (float*)(ws + 8759296);
    float* out1   = (float*)(ws + 16439296);
    float* out2   = (float*)(ws + 24119296);
    float* outs[3] = {out0, out1, out2};

    prep_kernel<<<1, 128, 0, stream>>>(h_index, t_index, r_index, h_set, r_set, inv_i, t_sel);
    lstm_kernel<<<1, 256, 0, stream>>>(qemb, wih, whh, bih, bhh, r_set,
                                       hstate, cstate, hidden, gates);

    const int NB = N_ENT * BN;                            // 1,920,000
    for (int step = 0; step < NSTEP; ++step) {
        attn_kernel<<<1, 128, 0, stream>>>(hidden, attn, step);
        inp_kernel<<<(NB + 255) / 256, 256, 0, stream>>>(attn, h_set, out0, out1, inp, step);
        wlog_kernel<<<30, 256, 0, stream>>>(hidden, weight_w, weight_b, wlog, step);
        wsoftmax_kernel<<<128, 256, 0, stream>>>(wlog, wsoft);
        zero_kernel<<<(NB + 255) / 256, 256, 0, stream>>>(outs[step], NB);
        edge_kernel<<<(E2 * 32 + 255) / 256, 256, 0, stream>>>(src, dst, rel, ew, inp,
                                                               wsoft, outs[step]);
        norm_kernel<<<128, 256, 0, stream>>>(outs[step]);
    }
    score_kernel<<<1, 128, 0, stream>>>(out2, t_sel, inv_i, lin_w, lin_b, (float*)d_out);
}